// QRes_40596030881873
// MI455X (gfx1250) — compile-verified
//
#include <hip/hip_runtime.h>
#include <math.h>

#define BQ   4096
#define SEQ  168
#define CF   64
#define PRED 96
#define KW   25
#define NQ   4
#define NLAG 11
#define NL   3
#define XST  72   // padded LDS row stride (floats): bank = (8l + c) mod 64

typedef __attribute__((ext_vector_type(2))) float        v2f;
typedef __attribute__((ext_vector_type(4))) float        v4f;
typedef __attribute__((ext_vector_type(8))) float        v8f;
typedef __attribute__((ext_vector_type(4))) unsigned int v4u;
typedef __attribute__((ext_vector_type(4))) int          v4i;
typedef __attribute__((ext_vector_type(8))) int          v8i;

__device__ __constant__ int d_lag[NLAG] = {167,166,165,164,163,162,145,144,143,1,0};

// ---------------- 4-qubit circuit simulation (per-thread, registers) ----------------
// State indexing: flat idx = q0*8 + q1*4 + q2*2 + q3  (qubit q -> bit (8 >> q))
__device__ __forceinline__ void run_circuit(const float* qin, const float* __restrict__ w,
                                            int pauli, float* out4)
{
    float sr[16], si[16];
#pragma unroll
    for (int i = 0; i < 16; ++i) { sr[i] = 0.f; si[i] = 0.f; }
    sr[0] = 1.f;

#pragma unroll
    for (int l = 0; l < NL; ++l) {
        // RY(qin[q]) on each qubit (real rotation)
#pragma unroll
        for (int q = 0; q < NQ; ++q) {
            float c = cosf(qin[q] * 0.5f);
            float s = sinf(qin[q] * 0.5f);
            int bit = 8 >> q;
#pragma unroll
            for (int i = 0; i < 16; ++i) {
                if (i & bit) continue;
                int j = i | bit;
                float a0r = sr[i], a0i = si[i], a1r = sr[j], a1i = si[j];
                sr[i] = c*a0r - s*a1r;  si[i] = c*a0i - s*a1i;
                sr[j] = s*a0r + c*a1r;  si[j] = s*a0i + c*a1i;
            }
        }
        // Rot(phi, theta, omega) per qubit
#pragma unroll
        for (int q = 0; q < NQ; ++q) {
            float phi = w[(l*NQ + q)*3 + 0];
            float th  = w[(l*NQ + q)*3 + 1];
            float om  = w[(l*NQ + q)*3 + 2];
            float ct = cosf(th*0.5f), st = sinf(th*0.5f);
            float A  = 0.5f*(phi + om), Bm = 0.5f*(phi - om);
            float ca = cosf(A),  sa = sinf(A);
            float cb = cosf(Bm), sb = sinf(Bm);
            float u00r =  ca*ct, u00i = -sa*ct;
            float u01r = -cb*st, u01i = -sb*st;
            float u10r =  cb*st, u10i = -sb*st;
            float u11r =  ca*ct, u11i =  sa*ct;
            int bit = 8 >> q;
#pragma unroll
            for (int i = 0; i < 16; ++i) {
                if (i & bit) continue;
                int j = i | bit;
                float a0r = sr[i], a0i = si[i], a1r = sr[j], a1i = si[j];
                sr[i] = u00r*a0r - u00i*a0i + u01r*a1r - u01i*a1i;
                si[i] = u00r*a0i + u00i*a0r + u01r*a1i + u01i*a1r;
                sr[j] = u10r*a0r - u10i*a0i + u11r*a1r - u11i*a1i;
                si[j] = u10r*a0i + u10i*a0r + u11r*a1i + u11i*a1r;
            }
        }
        // CNOT ring: control q, target (q+1)%4
#pragma unroll
        for (int q = 0; q < NQ; ++q) {
            int bc = 8 >> q;
            int bt = 8 >> ((q + 1) & 3);
#pragma unroll
            for (int i = 0; i < 16; ++i) {
                if ((i & bc) && !(i & bt)) {
                    int j = i | bt;
                    float tr = sr[i]; sr[i] = sr[j]; sr[j] = tr;
                    float ti = si[i]; si[i] = si[j]; si[j] = ti;
                }
            }
        }
    }
    // Expectation values
#pragma unroll
    for (int q = 0; q < NQ; ++q) {
        int bit = 8 >> q;
        float e = 0.f;
#pragma unroll
        for (int i = 0; i < 16; ++i) {
            if (i & bit) continue;
            int j = i | bit;
            if (pauli == 0)       e += 2.f * (sr[i]*sr[j] + si[i]*si[j]);   // X
            else if (pauli == 1)  e += 2.f * (sr[i]*si[j] - si[i]*sr[j]);   // Y
            else                  e += sr[i]*sr[i] + si[i]*si[i]
                                     - sr[j]*sr[j] - si[j]*si[j];           // Z
        }
        out4[q] = e;
    }
}

// ---- stage 1: one thread per (sample, circuit): 12288 threads ----
__global__ void __launch_bounds__(128)
qfeat_kernel(const float* __restrict__ x,
             const float* __restrict__ lagc_W, const float* __restrict__ lagc_b,
             const float* __restrict__ wx, const float* __restrict__ wy,
             const float* __restrict__ wz,
             float* __restrict__ feats)       // BQ * 12
{
    int t = blockIdx.x * blockDim.x + threadIdx.x;
    if (t >= BQ * 3) return;
    int b = t / 3, circ = t % 3;

    const float* xb = x + (size_t)b * SEQ * CF + (CF - 1);
    float lag[NLAG];
#pragma unroll
    for (int j = 0; j < NLAG; ++j) lag[j] = xb[d_lag[j] * CF];

    float qin[NQ];
#pragma unroll
    for (int q = 0; q < NQ; ++q) {
        float acc = lagc_b[q];
#pragma unroll
        for (int j = 0; j < NLAG; ++j) acc += lag[j] * lagc_W[q*NLAG + j];
        qin[q] = tanhf(acc) * 3.14159265358979f;
    }

    const float* w = (circ == 0) ? wx : ((circ == 1) ? wy : wz);
    float out4[4];
    run_circuit(qin, w, circ, out4);

    float* f = feats + (size_t)b * 12 + circ * 4;
    f[0] = out4[0]; f[1] = out4[1]; f[2] = out4[2]; f[3] = out4[3];
}

// ---- stage 2: one thread per (sample, p): 393216 threads ----
__global__ void __launch_bounds__(256)
qmlp_kernel(const float* __restrict__ feats,
            const float* __restrict__ W1, const float* __restrict__ b1,
            const float* __restrict__ W2, const float* __restrict__ b2,
            float* __restrict__ modout)       // BQ * PRED
{
    int t = blockIdx.x * blockDim.x + threadIdx.x;
    if (t >= BQ * PRED) return;
    int b = t / PRED, p = t % PRED;

    const float* f = feats + (size_t)b * 12;
    float fl[12];
#pragma unroll
    for (int j = 0; j < 12; ++j) fl[j] = f[j];

    float acc2 = b2[p];
#pragma unroll
    for (int i = 0; i < 12; ++i) {
        float hi = b1[i];
#pragma unroll
        for (int j = 0; j < 12; ++j) hi += fl[j] * W1[i*12 + j];
        hi = fmaxf(hi, 0.f);
        acc2 += hi * W2[p*12 + i];
    }
    modout[t] = tanhf(acc2);
}

// ---------------- main fused kernel: TDM stage + conv + dual WMMA GEMM + blend ----------------
__global__ void __launch_bounds__(256)
forecast_kernel(const float* __restrict__ x,
                const float* __restrict__ decomp_w,
                const float* __restrict__ trend_W, const float* __restrict__ trend_b,
                const float* __restrict__ seas_W,  const float* __restrict__ seas_b,
                const float* __restrict__ modv,    const float* __restrict__ alpha,
                float* __restrict__ out)
{
    __shared__ float sx [SEQ * XST];   // x tile, TDM-padded rows (48384 B)
    __shared__ float str[SEQ * XST];   // trend tile
    __shared__ float sker[CF * KW];    // softmax conv kernel (6400 B)

    const int tid = threadIdx.x;
    const int b   = blockIdx.x;
    const float* xb = x + (size_t)b * SEQ * CF;

    // ---- Stage x -> LDS via the Tensor Data Mover (wave 0 issues one 2D-tile DMA).
    // D# pad feature inserts 8 DWORDs after every 64 DWORDs -> LDS row stride = 72 floats,
    // so bank = (8*row + col) mod 64 and the two 16-lane halves of each WMMA B-fragment
    // read disjoint bank ranges.
    if ((tid >> 5) == 0) {
        unsigned lds_sx = (unsigned)(unsigned long long)
                          (__attribute__((address_space(3))) const void*)&sx[0];
        unsigned long long ga = (unsigned long long)xb;
        unsigned ga_lo = (unsigned)__builtin_amdgcn_readfirstlane((int)(ga & 0xffffffffu));
        unsigned ga_hi = (unsigned)__builtin_amdgcn_readfirstlane((int)(ga >> 32));

        v4u g0;
        g0.x = 1u;                                   // count=1 (valid descriptor)
        g0.y = lds_sx;                               // lds_addr
        g0.z = ga_lo;                                // global_addr[31:0]
        g0.w = (ga_hi & 0x01ffffffu) | (2u << 30);   // global_addr[56:32] | type=2

        v8i g1;
        g1[0] = (int)((2u << 16)        // data_size = 4 bytes
                    | (1u << 20)        // pad_enable
                    | (5u << 22)        // pad_interval: 64 DWORDs
                    | (7u << 25));      // pad_amount:   8 DWORDs
        g1[1] = (int)((unsigned)CF << 16);           // tensor_dim0 low16 = 64
        g1[2] = (int)((unsigned)SEQ << 16);          // dim0 hi16=0 | tensor_dim1 low16 = 168
        g1[3] = (int)((unsigned)CF << 16);           // dim1 hi16=0 | tile_dim0 = 64
        g1[4] = (int)(unsigned)SEQ;                  // tile_dim1 = 168, tile_dim2 = 0
        g1[5] = (int)(unsigned)CF;                   // tensor_dim0_stride low32 = 64
        g1[6] = 0;
        g1[7] = 0;

        v4i gz4 = {0, 0, 0, 0};
        v8i gz8 = {0, 0, 0, 0, 0, 0, 0, 0};
        __builtin_amdgcn_tensor_load_to_lds(g0, g1, gz4, gz4, gz8, 0);
    }

    // Per-channel softmax of decomp weights (overlaps with the DMA)
    if (tid < CF) {
        float w[KW]; float m = -1e30f;
#pragma unroll
        for (int k = 0; k < KW; ++k) { w[k] = decomp_w[tid*KW + k]; m = fmaxf(m, w[k]); }
        float sum = 0.f;
#pragma unroll
        for (int k = 0; k < KW; ++k) { w[k] = __expf(w[k] - m); sum += w[k]; }
        float inv = 1.f / sum;
#pragma unroll
        for (int k = 0; k < KW; ++k) sker[tid*KW + k] = w[k] * inv;
    }

    if ((tid >> 5) == 0) __builtin_amdgcn_s_wait_tensorcnt(0);
    __syncthreads();

    // Depthwise trend conv with edge padding -> LDS
    for (int idx = tid; idx < SEQ * CF; idx += 256) {
        int l = idx >> 6;
        int c = idx & 63;
        float acc = 0.f;
#pragma unroll
        for (int k = 0; k < KW; ++k) {
            int li = l + k - (KW - 1) / 2;
            li = min(max(li, 0), SEQ - 1);
            acc += sker[c*KW + k] * sx[li*XST + c];
        }
        str[l*XST + c] = acc;
    }
    __syncthreads();

    const int lane = tid & 31;
    const int wave = tid >> 5;
    const int half = lane >> 4;
    const int lid  = lane & 15;

    float aW = alpha[0];
    aW = 1.f / (1.f + __expf(-aW));          // sigmoid(alpha)
    const float  oneMa = 1.f - aW;
    const float* modb  = modv + (size_t)b * PRED;
    float* ob = out + (size_t)b * PRED * CF;

    // 24 output tiles (6 p-tiles x 4 c-tiles), 3 per wave, fully uniform (EXEC all 1s)
    for (int t = wave; t < 24; t += 8) {
        const int pb = (t % 6) * 16;
        const int cb = (t / 6) * 16;
        const int prow = pb + lid;           // A matrix: M = lid for both lane halves
        const int ccol = cb + lid;           // B/C matrices: N = lid for both halves

        v8f accT = {0.f,0.f,0.f,0.f,0.f,0.f,0.f,0.f};
        v8f accS = {0.f,0.f,0.f,0.f,0.f,0.f,0.f,0.f};

        for (int k = 0; k < SEQ; k += 4) {
            const int k0 = k + 2*half;       // lanes 0-15: K=k,k+1 ; lanes 16-31: K=k+2,k+3
            v2f aT, aS, bT, bS;
            aT.x = trend_W[prow*SEQ + k0];   aT.y = trend_W[prow*SEQ + k0 + 1];
            aS.x = seas_W [prow*SEQ + k0];   aS.y = seas_W [prow*SEQ + k0 + 1];
            float x0 = sx [ k0   *XST + ccol];
            float x1 = sx [(k0+1)*XST + ccol];
            float t0 = str[ k0   *XST + ccol];
            float t1 = str[(k0+1)*XST + ccol];
            bT.x = t0;      bT.y = t1;       // trend operand
            bS.x = x0 - t0; bS.y = x1 - t1;  // seasonal = x - trend, on the fly
            accT = __builtin_amdgcn_wmma_f32_16x16x4_f32(false, aT, false, bT,
                                                         (short)0, accT, false, false);
            accS = __builtin_amdgcn_wmma_f32_16x16x4_f32(false, aS, false, bS,
                                                         (short)0, accS, false, false);
        }

        // Epilogue: biases, modulation, alpha blend
#pragma unroll
        for (int r = 0; r < 8; ++r) {
            const int p = pb + r + 8*half;   // C/D layout: VGPR r -> M=r (lanes<16) / M=8+r
            float T = accT[r] + trend_b[p];
            float S = accS[r] + seas_b[p];
            float m = modb[p];
            ob[p*CF + ccol] = aW * S * (1.f + m) + oneMa * T;
        }
    }
}

extern "C" void kernel_launch(void* const* d_in, const int* in_sizes, int n_in,
                              void* d_out, int out_size, void* d_ws, size_t ws_size,
                              hipStream_t stream) {
    const float* x        = (const float*)d_in[0];
    const float* decomp_w = (const float*)d_in[1];
    const float* trend_W  = (const float*)d_in[2];
    const float* trend_b  = (const float*)d_in[3];
    const float* seas_W   = (const float*)d_in[4];
    const float* seas_b   = (const float*)d_in[5];
    const float* lagc_W   = (const float*)d_in[6];
    const float* lagc_b   = (const float*)d_in[7];
    const float* res_wx   = (const float*)d_in[8];
    const float* res_wy   = (const float*)d_in[9];
    const float* res_wz   = (const float*)d_in[10];
    const float* ro_W1    = (const float*)d_in[11];
    const float* ro_b1    = (const float*)d_in[12];
    const float* ro_W2    = (const float*)d_in[13];
    const float* ro_b2    = (const float*)d_in[14];
    const float* alpha    = (const float*)d_in[15];
    float* out   = (float*)d_out;
    float* featw = (float*)d_ws;                    // BQ*12 floats
    float* modw  = featw + (size_t)BQ * 12;         // BQ*PRED floats

    qfeat_kernel<<<(BQ*3 + 127)/128, 128, 0, stream>>>(x, lagc_W, lagc_b,
                                                       res_wx, res_wy, res_wz, featw);
    qmlp_kernel<<<(BQ*PRED)/256, 256, 0, stream>>>(featw, ro_W1, ro_b1, ro_W2, ro_b2, modw);
    forecast_kernel<<<BQ, 256, 0, stream>>>(x, decomp_w, trend_W, trend_b,
                                            seas_W, seas_b, modw, alpha, out);
}